// SlideGCD_ABMIL_43370579755063
// MI455X (gfx1250) — compile-verified
//
#include <hip/hip_runtime.h>
#include <hip/hip_bf16.h>

typedef __attribute__((ext_vector_type(2))) float v2f;
typedef __attribute__((ext_vector_type(8))) float v8f;

#define BATCH  16
#define N_INST 2048
#define FEAT   1024
#define LDIM   500
#define DDIM   128
#define CDIM   2
#define KNN    12
#define BUFN   512
#define MAXN   528
#define MROWS  (BATCH * N_INST)   // 32768

// ---------------------------------------------------------------------------
// WMMA GEMM: Out[M,NCOLS] = relu(A[M,KDIM] @ B[KDIM,NCOLS] + bias)
// fp32 V_WMMA_F32_16X16X4_F32 (exact fp32 vs reference).
// Block tile 128x64, 8 waves, each wave = one 16x64 strip (4x v8f accum).
// Double-buffered LDS panels; next panel's global loads overlap current WMMAs.
// ---------------------------------------------------------------------------
template<int KDIM, int NCOLS>
__global__ __launch_bounds__(256)
void gemm_bias_relu_wmma(const float* __restrict__ A, const float* __restrict__ B,
                         const float* __restrict__ bias, float* __restrict__ Out)
{
    constexpr int KB = (KDIM + 31) / 32;
    // padded to reduce LDS bank conflicts (stride 132/68 v2f = 264/136 dwords)
    __shared__ v2f As[2][16][132];   // As[buf][k/2][row] = (A[row][2k'], A[row][2k'+1])
    __shared__ v2f Bs[2][16][68];    // Bs[buf][k/2][n]   = (B[2k'][n],  B[2k'+1][n])

    const int tid  = threadIdx.x;
    const int lane = tid & 31;
    const int wave = tid >> 5;
    const int mBlock = blockIdx.x * 128;
    const int nBlock = blockIdx.y * 64;

    v8f acc[4] = {};

    // staging registers: A = 4x float4 (128 rows x 32 k / 256 thr), B = 8 scalars
    float4 ra[4];
    float  rb[8];

    // per-thread A mapping: c4 = tid&7 (which float4 in k), rows tid>>3 + {0,32,64,96}
    const int a_c4  = tid & 7;
    const int a_row = tid >> 3;
    // per-thread B mapping: n = tid&63, k = tid>>6 + {0,4,...,28}
    const int b_n = tid & 63;
    const int b_k = tid >> 6;

    auto loadPanel = [&](int kb) {
        const int k0 = kb * 32;
        const bool full = (k0 + 32 <= KDIM);
        #pragma unroll
        for (int i = 0; i < 4; ++i) {
            int row = a_row + i * 32;
            const float* p = A + (size_t)(mBlock + row) * KDIM + k0 + a_c4 * 4;
            if (full) {
                ra[i] = *(const float4*)p;
            } else {
                int kb4 = k0 + a_c4 * 4;
                float4 v;
                v.x = (kb4 + 0 < KDIM) ? p[0] : 0.f;
                v.y = (kb4 + 1 < KDIM) ? p[1] : 0.f;
                v.z = (kb4 + 2 < KDIM) ? p[2] : 0.f;
                v.w = (kb4 + 3 < KDIM) ? p[3] : 0.f;
                ra[i] = v;
            }
        }
        #pragma unroll
        for (int i = 0; i < 8; ++i) {
            int k = b_k + i * 4;
            float v = 0.f;
            if ((k0 + k < KDIM) && (nBlock + b_n < NCOLS))
                v = B[(size_t)(k0 + k) * NCOLS + nBlock + b_n];
            rb[i] = v;
        }
    };
    auto storePanel = [&](int buf) {
        #pragma unroll
        for (int i = 0; i < 4; ++i) {
            int row = a_row + i * 32;
            v2f lo; lo.x = ra[i].x; lo.y = ra[i].y;
            v2f hi; hi.x = ra[i].z; hi.y = ra[i].w;
            As[buf][a_c4 * 2 + 0][row] = lo;
            As[buf][a_c4 * 2 + 1][row] = hi;
        }
        #pragma unroll
        for (int i = 0; i < 8; ++i) {
            int k = b_k + i * 4;
            ((float*)&Bs[buf][k >> 1][b_n])[k & 1] = rb[i];
        }
    };

    loadPanel(0);
    storePanel(0);

    const int half = lane >> 4;   // selects K pair (k vs k+2)
    const int l16  = lane & 15;

    for (int kb = 0; kb < KB; ++kb) {
        __syncthreads();                      // panel `cur` visible to all waves
        const int cur = kb & 1;
        if (kb + 1 < KB) loadPanel(kb + 1);   // HBM latency hidden under WMMAs
        #pragma unroll
        for (int kk = 0; kk < 32; kk += 4) {
            int p = (kk >> 1) + half;
            v2f afrag = As[cur][p][wave * 16 + l16];
            #pragma unroll
            for (int t = 0; t < 4; ++t) {
                v2f bfrag = Bs[cur][p][t * 16 + l16];
                acc[t] = __builtin_amdgcn_wmma_f32_16x16x4_f32(
                    false, afrag, false, bfrag, (short)0, acc[t], false, false);
            }
        }
        if (kb + 1 < KB) storePanel(cur ^ 1);
    }

    // Epilogue: bias + ReLU, mask padded columns.
    #pragma unroll
    for (int t = 0; t < 4; ++t) {
        int gn = nBlock + t * 16 + l16;
        if (gn >= NCOLS) continue;
        float bs = bias[gn];
        #pragma unroll
        for (int r = 0; r < 8; ++r) {
            int gm  = mBlock + wave * 16 + r + half * 8;
            float v = acc[t][r] + bs;
            Out[(size_t)gm * NCOLS + gn] = v > 0.f ? v : 0.f;
        }
    }
}

// a[i] = s[i,:] . Wv + bv   (s = relu(h@Wa+ba), [32768,128])
__global__ __launch_bounds__(256)
void attn_score(const float* __restrict__ s, const float* __restrict__ Wv,
                const float* __restrict__ bv, float* __restrict__ a)
{
    int i = blockIdx.x * 256 + threadIdx.x;
    if (i >= MROWS) return;
    float acc = bv[0];
    const float* sr = s + (size_t)i * DDIM;
    #pragma unroll 4
    for (int k = 0; k < DDIM; ++k) acc += sr[k] * Wv[k];
    a[i] = acc;
}

// Per bag: softmax over instances, emb = A . h, logits_mlp = emb@Wc+bc
__global__ __launch_bounds__(256)
void softmax_emb(const float* __restrict__ a, const float* __restrict__ h,
                 const float* __restrict__ Wc, const float* __restrict__ bc,
                 float* __restrict__ xc, float* __restrict__ out_logits)
{
    __shared__ float ab[N_INST];
    __shared__ float red[256];
    __shared__ float emb[LDIM];
    const int b = blockIdx.x, tid = threadIdx.x;
    const float* ag = a + b * N_INST;

    float lmax = -1e30f;
    for (int n = tid; n < N_INST; n += 256) { float v = ag[n]; ab[n] = v; lmax = fmaxf(lmax, v); }
    red[tid] = lmax; __syncthreads();
    for (int st = 128; st > 0; st >>= 1) { if (tid < st) red[tid] = fmaxf(red[tid], red[tid + st]); __syncthreads(); }
    float mx = red[0]; __syncthreads();

    float lsum = 0.f;
    for (int n = tid; n < N_INST; n += 256) { float e = expf(ab[n] - mx); ab[n] = e; lsum += e; }
    red[tid] = lsum; __syncthreads();
    for (int st = 128; st > 0; st >>= 1) { if (tid < st) red[tid] += red[tid + st]; __syncthreads(); }
    float inv = 1.f / red[0];
    __syncthreads();

    for (int l = tid; l < LDIM; l += 256) {
        float acc = 0.f;
        const float* hb = h + (size_t)b * N_INST * LDIM + l;
        for (int n = 0; n < N_INST; ++n) acc += ab[n] * hb[(size_t)n * LDIM];
        acc *= inv;
        emb[l] = acc;
        xc[b * LDIM + l] = acc;
    }
    __syncthreads();
    if (tid < CDIM) {
        float acc = bc[tid];
        for (int l = 0; l < LDIM; ++l) acc += emb[l] * Wc[l * CDIM + tid];
        out_logits[b * CDIM + tid] = acc;
    }
}

__global__ void copy_rehearsal(const float* __restrict__ reh, float* __restrict__ xc)
{
    int i = blockIdx.x * 256 + threadIdx.x;
    if (i < BUFN * LDIM) xc[BATCH * LDIM + i] = reh[i];
}

// 8 rows of X per block: Out[i,:] = act(X[i,:]@W + b)
// MODE 0: leaky(0.2) + row L2-normalize; MODE 1: relu
template<int MODE>
__global__ __launch_bounds__(256)
void rowgemm8(const float* __restrict__ X, const float* __restrict__ W,
              const float* __restrict__ bias, float* __restrict__ Out)
{
    __shared__ float xr[8][LDIM];
    __shared__ float zr[8][LDIM];
    const int i0 = blockIdx.x * 8, tid = threadIdx.x;

    for (int e = tid; e < 8 * LDIM; e += 256) xr[e / LDIM][e % LDIM] = X[(size_t)i0 * LDIM + e];
    __syncthreads();

    for (int j = tid; j < LDIM; j += 256) {
        float acc[8];
        #pragma unroll
        for (int r = 0; r < 8; ++r) acc[r] = 0.f;
        #pragma unroll 4
        for (int l = 0; l < LDIM; ++l) {
            float wv = W[(size_t)l * LDIM + j];   // one load feeds 8 FMAs
            #pragma unroll
            for (int r = 0; r < 8; ++r) acc[r] += xr[r][l] * wv;
        }
        float bj = bias[j];
        #pragma unroll
        for (int r = 0; r < 8; ++r) {
            float v = acc[r] + bj;
            if (MODE == 0) v = v > 0.f ? v : 0.2f * v;
            else           v = v > 0.f ? v : 0.f;
            zr[r][j] = v;
        }
    }
    __syncthreads();

    if (MODE == 0) {
        // one wave per row: L2 norm via shuffle butterfly
        const int w = tid >> 5, lane = tid & 31;
        float ls = 0.f;
        for (int l = lane; l < LDIM; l += 32) { float v = zr[w][l]; ls += v * v; }
        #pragma unroll
        for (int off = 16; off > 0; off >>= 1) ls += __shfl_xor(ls, off, 32);
        float inv = 1.f / (sqrtf(ls) + 1e-8f);
        for (int l = lane; l < LDIM; l += 32) Out[(size_t)(i0 + w) * LDIM + l] = zr[w][l] * inv;
    } else {
        for (int e = tid; e < 8 * LDIM; e += 256) Out[(size_t)i0 * LDIM + e] = zr[e / LDIM][e % LDIM];
    }
}

// 8 sim rows per block + per-row top-12 (stable desc, low index ties) + edge softmax
__global__ __launch_bounds__(256)
void sim_topk8(const float* __restrict__ zn, float* __restrict__ wbuf, int* __restrict__ ibuf)
{
    __shared__ float zi[8][LDIM];
    __shared__ float simr[8][MAXN];
    const int i0 = blockIdx.x * 8, tid = threadIdx.x;

    for (int e = tid; e < 8 * LDIM; e += 256) zi[e / LDIM][e % LDIM] = zn[(size_t)i0 * LDIM + e];
    __syncthreads();

    for (int j = tid; j < MAXN; j += 256) {
        float acc[8];
        #pragma unroll
        for (int r = 0; r < 8; ++r) acc[r] = 0.f;
        const float* zj = zn + (size_t)j * LDIM;
        #pragma unroll 4
        for (int l = 0; l < LDIM; ++l) {
            float v = zj[l];
            #pragma unroll
            for (int r = 0; r < 8; ++r) acc[r] += zi[r][l] * v;
        }
        #pragma unroll
        for (int r = 0; r < 8; ++r) simr[r][j] = acc[r];
    }
    __syncthreads();

    if (tid < 8) {
        const int i = i0 + tid;
        float tv[KNN]; int ti[KNN];
        #pragma unroll
        for (int k = 0; k < KNN; ++k) { tv[k] = -1e30f; ti[k] = 0; }
        for (int j = 0; j < MAXN; ++j) {
            float v = simr[tid][j];
            if (v > tv[KNN - 1]) {
                int p = KNN - 1;
                while (p > 0 && v > tv[p - 1]) { tv[p] = tv[p - 1]; ti[p] = ti[p - 1]; --p; }
                tv[p] = v; ti[p] = j;
            }
        }
        float mx = tv[0], sum = 0.f, e[KNN];
        #pragma unroll
        for (int k = 0; k < KNN; ++k) { e[k] = expf(tv[k] - mx); sum += e[k]; }
        float inv = 1.f / sum;
        #pragma unroll
        for (int k = 0; k < KNN; ++k) { wbuf[i * KNN + k] = e[k] * inv; ibuf[i * KNN + k] = ti[k]; }
    }
}

// m[i,:] = sum_k w[i,k] * feat[idx[i,k], :]
__global__ __launch_bounds__(256)
void gather_msg(const float* __restrict__ feat, const float* __restrict__ wbuf,
                const int* __restrict__ ibuf, float* __restrict__ Out)
{
    __shared__ float wk[KNN];
    __shared__ int   ik[KNN];
    const int i = blockIdx.x, tid = threadIdx.x;
    if (tid < KNN) { wk[tid] = wbuf[i * KNN + tid]; ik[tid] = ibuf[i * KNN + tid]; }
    __syncthreads();
    for (int l = tid; l < LDIM; l += 256) {
        float acc = 0.f;
        #pragma unroll
        for (int k = 0; k < KNN; ++k) acc += wk[k] * feat[(size_t)ik[k] * LDIM + l];
        Out[(size_t)i * LDIM + l] = acc;
    }
}

__global__ void graph_logits(const float* __restrict__ m2, const float* __restrict__ Wg2,
                             const float* __restrict__ bg2, float* __restrict__ out)
{
    int t = threadIdx.x;
    if (t >= BATCH * CDIM) return;
    int b = t / CDIM, c = t % CDIM;
    float acc = bg2[c];
    for (int l = 0; l < LDIM; ++l) acc += m2[(size_t)b * LDIM + l] * Wg2[l * CDIM + c];
    out[BATCH * CDIM + t] = acc;
}

extern "C" void kernel_launch(void* const* d_in, const int* in_sizes, int n_in,
                              void* d_out, int out_size, void* d_ws, size_t ws_size,
                              hipStream_t stream)
{
    const float* x    = (const float*)d_in[0];
    const float* W1   = (const float*)d_in[1];
    const float* b1   = (const float*)d_in[2];
    const float* Wa   = (const float*)d_in[3];
    const float* ba   = (const float*)d_in[4];
    const float* Wv   = (const float*)d_in[5];
    const float* bv   = (const float*)d_in[6];
    const float* Wc   = (const float*)d_in[7];
    const float* bc   = (const float*)d_in[8];
    const float* reh  = (const float*)d_in[9];
    const float* Wg   = (const float*)d_in[10];
    const float* bg   = (const float*)d_in[11];
    const float* Wg1  = (const float*)d_in[12];
    const float* bg1  = (const float*)d_in[13];
    const float* Wg2  = (const float*)d_in[14];
    const float* bg2  = (const float*)d_in[15];
    float* out = (float*)d_out;

    float* ws = (float*)d_ws;
    // workspace layout (float elements)
    float* h    = ws;                                   // 32768*500
    float* s    = ws + (size_t)16384000;                // 32768*128
    float* a    = ws + (size_t)20578304;                // 32768
    float* xc   = ws + (size_t)20611072;                // 528*500
    float* zn   = ws + (size_t)20875072;                // 528*500
    float* m1   = ws + (size_t)21139072;                // 528*500
    float* h1   = ws + (size_t)21403072;                // 528*500
    float* m2   = ws + (size_t)21667072;                // 528*500
    float* wbuf = ws + (size_t)21931072;                // 528*12
    int*   ibuf = (int*)(ws + (size_t)21937408);        // 528*12

    // 1) h = relu(x @ W1 + b1)  — 33.6 GFLOP fp32 WMMA GEMM (double-buffered)
    gemm_bias_relu_wmma<FEAT, LDIM><<<dim3(MROWS / 128, 8), 256, 0, stream>>>(x, W1, b1, h);
    // 2) s = relu(h @ Wa + ba)
    gemm_bias_relu_wmma<LDIM, DDIM><<<dim3(MROWS / 128, 2), 256, 0, stream>>>(h, Wa, ba, s);
    // 3) a = s @ Wv + bv
    attn_score<<<MROWS / 256, 256, 0, stream>>>(s, Wv, bv, a);
    // 4) per-bag softmax + attention pooling + MLP logits (out[0:32])
    softmax_emb<<<BATCH, 256, 0, stream>>>(a, h, Wc, bc, xc, out);
    // 5) rehearsal rows into xc[16:528]
    copy_rehearsal<<<(BUFN * LDIM + 255) / 256, 256, 0, stream>>>(reh, xc);
    // 6) zn = normalize(leaky_relu(xc @ Wg + bg)), 8 rows/block
    rowgemm8<0><<<MAXN / 8, 256, 0, stream>>>(xc, Wg, bg, zn);
    // 7) cosine kNN (top-12) + edge softmax, 8 rows/block
    sim_topk8<<<MAXN / 8, 256, 0, stream>>>(zn, wbuf, ibuf);
    // 8) m1 = weighted neighbor sum of xc
    gather_msg<<<MAXN, 256, 0, stream>>>(xc, wbuf, ibuf, m1);
    // 9) h1 = relu(m1 @ Wg1 + bg1), 8 rows/block
    rowgemm8<1><<<MAXN / 8, 256, 0, stream>>>(m1, Wg1, bg1, h1);
    // 10) m2 = weighted neighbor sum of h1
    gather_msg<<<MAXN, 256, 0, stream>>>(h1, wbuf, ibuf, m2);
    // 11) logits_graph (out[32:64])
    graph_logits<<<1, 32, 0, stream>>>(m2, Wg2, bg2, out);
}